// BarrierNet_36979668418812
// MI455X (gfx1250) — compile-verified
//
#include <hip/hip_runtime.h>
#include <hip/hip_bf16.h>
#include <math.h>

typedef __attribute__((ext_vector_type(2))) float v2f;
typedef __attribute__((ext_vector_type(8))) float v8f;

#define HID_STRIDE 132   // 128 + 4 pad: avoids 16-way LDS bank conflicts on column writes
#define X2_STRIDE  33    // 32 + 1 pad
#define ROWS_BLK   64    // batch rows per block (4 waves x 16-row WMMA tiles)

__global__ __launch_bounds__(128) void bnet_mlp_kernel(
    const float* __restrict__ x,
    const float* __restrict__ mean, const float* __restrict__ stdv,
    const float* __restrict__ W1,  const float* __restrict__ b1,
    const float* __restrict__ W21, const float* __restrict__ b21,
    const float* __restrict__ W22, const float* __restrict__ b22,
    const float* __restrict__ W31, const float* __restrict__ b31,
    const float* __restrict__ W32, const float* __restrict__ b32,
    float* __restrict__ aux, float* __restrict__ out, int nObs)
{
    __shared__ float s_hid[ROWS_BLK * HID_STRIDE];  // 33,792 B
    __shared__ float s_x21[ROWS_BLK * X2_STRIDE];   //  8,448 B
    __shared__ float s_x22[ROWS_BLK * X2_STRIDE];   //  8,448 B

    const int tid  = threadIdx.x;
    const int lane = tid & 31;
    const int wave = tid >> 5;          // 0..3
    const int m    = lane & 15;         // row-in-tile (A) or col-in-tile (B/C)
    const int kh   = lane >> 4;         // K-half select (0/1)
    const int rowBase = blockIdx.x * ROWS_BLK + wave * 16;

    // ---------------- GEMM1: hid = relu(x @ W1 + b1), raw x, K=8 ----------------
    v8f c1[8];
    #pragma unroll
    for (int t = 0; t < 8; ++t) c1[t] = (v8f){0,0,0,0,0,0,0,0};

    #pragma unroll
    for (int kk = 0; kk < 2; ++kk) {
        const int kb = 4*kk + 2*kh;                       // K = kb, kb+1 for this lane
        const float* xr = x + (size_t)(rowBase + m) * 8 + kb;
        v2f a; a.x = xr[0]; a.y = xr[1];                  // 8B-aligned pair
        #pragma unroll
        for (int t = 0; t < 8; ++t) {
            v2f b; b.x = W1[kb*128 + 16*t + m];
                   b.y = W1[(kb+1)*128 + 16*t + m];
            c1[t] = __builtin_amdgcn_wmma_f32_16x16x4_f32(
                        false, a, false, b, (short)0, c1[t], false, false);
        }
    }
    #pragma unroll
    for (int t = 0; t < 8; ++t) {
        const int col = 16*t + m;
        const float bias = b1[col];
        #pragma unroll
        for (int g = 0; g < 8; ++g) {
            float hv = fmaxf(c1[t][g] + bias, 0.0f);
            s_hid[(wave*16 + g + 8*kh) * HID_STRIDE + col] = hv;
        }
    }
    __syncthreads();

    // ---------------- GEMM2: x21/x22 = relu(hid @ W2x + b2x), K=128 ----------------
    v8f c21[2], c22[2];
    #pragma unroll
    for (int t = 0; t < 2; ++t) { c21[t] = (v8f){0,0,0,0,0,0,0,0};
                                  c22[t] = (v8f){0,0,0,0,0,0,0,0}; }
    for (int k = 0; k < 32; ++k) {
        const int kb = 4*k + 2*kh;
        v2f a = *(const v2f*)&s_hid[(wave*16 + m) * HID_STRIDE + kb]; // even offset: 8B aligned
        #pragma unroll
        for (int t = 0; t < 2; ++t) {
            v2f b;  b.x  = W21[kb*32 + 16*t + m];  b.y  = W21[(kb+1)*32 + 16*t + m];
            c21[t] = __builtin_amdgcn_wmma_f32_16x16x4_f32(
                         false, a, false, b, (short)0, c21[t], false, false);
            v2f b2; b2.x = W22[kb*32 + 16*t + m];  b2.y = W22[(kb+1)*32 + 16*t + m];
            c22[t] = __builtin_amdgcn_wmma_f32_16x16x4_f32(
                         false, a, false, b2, (short)0, c22[t], false, false);
        }
    }
    #pragma unroll
    for (int t = 0; t < 2; ++t) {
        const int col = 16*t + m;
        const float bb21 = b21[col], bb22 = b22[col];
        #pragma unroll
        for (int g = 0; g < 8; ++g) {
            const int r = wave*16 + g + 8*kh;
            s_x21[r * X2_STRIDE + col] = fmaxf(c21[t][g] + bb21, 0.0f);
            s_x22[r * X2_STRIDE + col] = fmaxf(c22[t][g] + bb22, 0.0f);
        }
    }
    __syncthreads();

    // ---------------- Heads + opponent barrier row + aux, one thread per row ----------------
    if (tid < ROWS_BLK) {
        const int r = tid;
        const int grow = blockIdx.x * ROWS_BLK + r;
        float x0[8];
        #pragma unroll
        for (int i = 0; i < 8; ++i)
            x0[i] = x[(size_t)grow*8 + i] * stdv[i] + mean[i];

        float z0 = b31[0], z1 = b31[1], w0 = b32[0], w1 = b32[1];
        #pragma unroll 4
        for (int j = 0; j < 32; ++j) {
            const float a21 = s_x21[r * X2_STRIDE + j];
            const float a22 = s_x22[r * X2_STRIDE + j];
            z0 += a21 * W31[2*j];  z1 += a21 * W31[2*j+1];
            w0 += a22 * W32[2*j];  w1 += a22 * W32[2*j+1];
        }
        const float p0 = 4.0f / (1.0f + expf(-w0));
        const float p1 = 4.0f / (1.0f + expf(-w1));
        const float psum = p0 + p1, pprod = p0 * p1;

        const float px = x0[0], py = x0[1], th = x0[2], v = x0[3];
        const float s = sinf(th), c = cosf(th);
        const float vs = v * s, vc = v * c;

        float* A = aux + (size_t)grow * 8;
        A[0]=px; A[1]=py; A[2]=vs; A[3]=vc; A[4]=s; A[5]=c; A[6]=psum; A[7]=pprod;

        // opponent row (index nObs)
        const float opx = x0[4], opy = x0[5], oth = x0[6], ov = x0[7];
        const float dxo = px - opx, dyo = py - opy;
        const float Ro = 1.1f;                                   // 2*0.5 + 0.1
        const float ob  = dxo*dxo + dyo*dyo - Ro*Ro;
        const float obd = 2.0f*(dxo*(vc - ov*cosf(oth)) + dyo*(vs - ov*sinf(oth)));
        const float oLf2b = 2.0f*(v*v + ov*ov - 2.0f*v*ov*cosf(th + oth));
        const float G0 = 2.0f*(dxo*vs - dyo*vc);                 // -LgLfbu1
        const float G1 = -2.0f*(dxo*c + dyo*s);                  // -LgLfbu2
        const float ho = oLf2b + psum*obd + pprod*ob;

        const size_t stride = (size_t)(3*nObs + 5);
        float* O = out + (size_t)grow * stride;
        O[0] = z0; O[1] = z1;
        O[2 + 2*nObs]     = G0;
        O[2 + 2*nObs + 1] = G1;
        O[2 + 2*(nObs+1) + nObs] = ho;
    }
}

__global__ __launch_bounds__(256) void bnet_barrier_kernel(
    const float* __restrict__ aux, const float* __restrict__ obst,
    float* __restrict__ out, int nObs)
{
    const int o = threadIdx.x;                // one lane per obstacle, held in registers
    const float ox = obst[3*o + 0];
    const float oy = obst[3*o + 1];
    const float R  = obst[3*o + 2] + 0.6f;    // AGENT_RADIUS + SAFETY_DIST + r
    const float R2 = R * R;

    __shared__ float s_aux[ROWS_BLK * 8];
    const int rbase = blockIdx.x * ROWS_BLK;
    for (int i = threadIdx.x; i < ROWS_BLK * 8; i += blockDim.x)
        s_aux[i] = aux[(size_t)rbase * 8 + i];
    __syncthreads();

    const size_t stride = (size_t)(3*nObs + 5);
    for (int r = 0; r < ROWS_BLK; ++r) {
        const float* A = &s_aux[r * 8];
        const float px = A[0], py = A[1], vs = A[2], vc = A[3];
        const float s  = A[4], c  = A[5], psum = A[6], pprod = A[7];

        const float dx = px - ox, dy = py - oy;
        const float barrier = dx*dx + dy*dy - R2;
        const float bdot = 2.0f*(dx*vc + dy*vs);
        const float G0   = 2.0f*(dx*vs - dy*vc);
        const float G1   = -2.0f*(dx*c + dy*s);
        const float Lf2b = 2.0f*(vs*vs + vc*vc);   // == 2*v^2
        const float h = Lf2b + psum*bdot + pprod*barrier;

        float* O = out + (size_t)(rbase + r) * stride;
        O[2 + 2*o]     = G0;                 // lane-contiguous pair stores
        O[2 + 2*o + 1] = G1;
        O[2 + 2*(nObs+1) + o] = h;           // lane-contiguous b32 stores
    }
}

extern "C" void kernel_launch(void* const* d_in, const int* in_sizes, int n_in,
                              void* d_out, int out_size, void* d_ws, size_t ws_size,
                              hipStream_t stream) {
    const float* x    = (const float*)d_in[0];
    const float* obst = (const float*)d_in[1];
    const float* mean = (const float*)d_in[2];
    const float* stdv = (const float*)d_in[3];
    const float* W1   = (const float*)d_in[4];  const float* b1  = (const float*)d_in[5];
    const float* W21  = (const float*)d_in[6];  const float* b21 = (const float*)d_in[7];
    const float* W22  = (const float*)d_in[8];  const float* b22 = (const float*)d_in[9];
    const float* W31  = (const float*)d_in[10]; const float* b31 = (const float*)d_in[11];
    const float* W32  = (const float*)d_in[12]; const float* b32 = (const float*)d_in[13];

    const int B    = in_sizes[0] / 8;
    const int nObs = in_sizes[1] / 3;   // 256
    float* aux = (float*)d_ws;          // B*8 floats = 2 MB
    float* out = (float*)d_out;

    const int nBlocks = B / ROWS_BLK;
    bnet_mlp_kernel<<<nBlocks, 128, 0, stream>>>(x, mean, stdv, W1, b1, W21, b21,
                                                 W22, b22, W31, b31, W32, b32,
                                                 aux, out, nObs);
    bnet_barrier_kernel<<<nBlocks, nObs, 0, stream>>>(aux, obst, out, nObs);
}